// AttentionBasedPredictor_18511309046070
// MI455X (gfx1250) — compile-verified
//
#include <hip/hip_runtime.h>
#include <math.h>

#define DIM_Q 512
#define DIM_G 128
#define KTOP 10
#define NB_SIM 512          // blocks for the sim/top-k kernel
#define WAVES_PER_BLOCK 8
#define TILE_ROWS (WAVES_PER_BLOCK * 16)   // 128 rows per block tile

typedef __attribute__((ext_vector_type(2))) float v2f;
typedef __attribute__((ext_vector_type(8))) float v8f;

// ---------------------------------------------------------------------------
// Kernel 1: pred_query[g] = dot(qt_hat, W_q[g,:]) + b_q[g]   (tiny)
// ---------------------------------------------------------------------------
__global__ __launch_bounds__(128) void pq_kernel(
    const float* __restrict__ qt, const float* __restrict__ Wq,
    const float* __restrict__ bq, float* __restrict__ pq) {
    __shared__ float q[DIM_Q];
    for (int i = threadIdx.x; i < DIM_Q; i += 128) q[i] = qt[i];
    __syncthreads();
    const int g = threadIdx.x;
    if (g < DIM_G) {
        float acc = bq[g];
        const float* w = Wq + (size_t)g * DIM_Q;
        #pragma unroll 8
        for (int d = 0; d < DIM_Q; ++d) acc = fmaf(q[d], w[d], acc);
        pq[g] = acc;
    }
}

// ---------------------------------------------------------------------------
// Kernel 2: sim = pred_query @ memory_key.T via V_WMMA_F32_16X16X4_F32,
// fused per-block top-10 reduction. Each wave handles 16 rows per tile.
// A fragment = 16 key rows (16x4), B fragment = pred_query chunk broadcast
// across all 16 columns, so D[m][*] = sim[row0+m].
// ---------------------------------------------------------------------------
__global__ __launch_bounds__(256) void sim_topk_kernel(
    const float* __restrict__ key, const float* __restrict__ pq_g,
    float* __restrict__ cand_val, int* __restrict__ cand_idx, int M) {

    __shared__ float pq[DIM_G];
    for (int i = threadIdx.x; i < DIM_G; i += 256) pq[i] = pq_g[i];
    __syncthreads();

    const int lane = threadIdx.x & 31;
    const int wave = threadIdx.x >> 5;
    const int hf   = lane >> 4;     // 0: K=0,1  1: K=2,3  (A-matrix 16x4 layout)
    const int m16  = lane & 15;

    // B fragments: B[k][n] = pq[k0+k] for all n  ->  per ISA layout
    // VGPR0 = rows {0,2}, VGPR1 = rows {1,3}  => b = {pq[k0+2*hf], pq[k0+2*hf+1]}
    v2f bfr[DIM_G / 4];
    #pragma unroll
    for (int kc = 0; kc < DIM_G / 4; ++kc) {
        bfr[kc].x = pq[kc * 4 + 2 * hf];
        bfr[kc].y = pq[kc * 4 + 2 * hf + 1];
    }

    // per-thread top-10 (sorted descending), only lanes 0/16 ever insert
    float tv[KTOP]; int ti[KTOP];
    #pragma unroll
    for (int j = 0; j < KTOP; ++j) { tv[j] = -__builtin_inff(); ti[j] = 0; }

    const int numTiles = (M + TILE_ROWS - 1) / TILE_ROWS;
    for (int t = blockIdx.x; t < numTiles; t += gridDim.x) {
        const int r0 = t * TILE_ROWS + wave * 16;
        int r = r0 + m16;
        if (r >= M) r = M - 1;                       // clamp; discarded later
        const float* rowp = key + (size_t)r * DIM_G + 2 * hf;

        v8f c = {};
        #pragma unroll
        for (int kc = 0; kc < DIM_G / 4; ++kc) {
            v2f a = *(const v2f*)(rowp + kc * 4);    // global_load_b64
            c = __builtin_amdgcn_wmma_f32_16x16x4_f32(
                    false, a, false, bfr[kc], (short)0, c, false, false);
        }

        // D layout: VGPR j -> lanes 0-15: M=j (N=lane), lanes 16-31: M=j+8.
        // All columns equal; lane 0 owns rows r0+0..7, lane 16 owns r0+8..15.
        if (m16 == 0) {
            const int base = r0 + (hf << 3);
            #pragma unroll
            for (int j = 0; j < 8; ++j) {
                const int row = base + j;
                const float v = c[j];
                if (row < M && v > tv[KTOP - 1]) {
                    int p = KTOP - 1;
                    #pragma unroll
                    for (int q2 = KTOP - 2; q2 >= 0; --q2) {
                        if (v > tv[q2]) { tv[q2 + 1] = tv[q2]; ti[q2 + 1] = ti[q2]; p = q2; }
                    }
                    tv[p] = v; ti[p] = row;
                }
            }
        }
    }

    // ---- block-level top-10 merge over 256 per-thread lists ----
    __shared__ float cv[256 * KTOP];
    __shared__ int   ci[256 * KTOP];
    __shared__ float rv[256];
    __shared__ int   rp[256];
    __shared__ float outv[KTOP];
    __shared__ int   outi[KTOP];

    #pragma unroll
    for (int j = 0; j < KTOP; ++j) {
        cv[threadIdx.x * KTOP + j] = tv[j];
        ci[threadIdx.x * KTOP + j] = ti[j];
    }
    __syncthreads();

    for (int sel = 0; sel < KTOP; ++sel) {
        float bv = cv[threadIdx.x * KTOP];
        int   bp = threadIdx.x * KTOP;
        #pragma unroll
        for (int j = 1; j < KTOP; ++j) {
            const float v = cv[threadIdx.x * KTOP + j];
            if (v > bv) { bv = v; bp = threadIdx.x * KTOP + j; }
        }
        rv[threadIdx.x] = bv; rp[threadIdx.x] = bp;
        __syncthreads();
        for (int s = 128; s > 0; s >>= 1) {
            if (threadIdx.x < s) {
                if (rv[threadIdx.x + s] > rv[threadIdx.x]) {
                    rv[threadIdx.x] = rv[threadIdx.x + s];
                    rp[threadIdx.x] = rp[threadIdx.x + s];
                }
            }
            __syncthreads();
        }
        if (threadIdx.x == 0) {
            outv[sel] = rv[0];
            outi[sel] = ci[rp[0]];
            cv[rp[0]] = -__builtin_inff();
        }
        __syncthreads();
    }

    if (threadIdx.x < KTOP) {
        cand_val[blockIdx.x * KTOP + threadIdx.x] = outv[threadIdx.x];
        cand_idx[blockIdx.x * KTOP + threadIdx.x] = outi[threadIdx.x];
    }
}

// ---------------------------------------------------------------------------
// Kernel 3: merge NB_SIM*10 candidates -> global top-10, softmax, gather
// memory_value rows, dot with pred_query, sigmoid. Single block.
// ---------------------------------------------------------------------------
__global__ __launch_bounds__(256) void final_kernel(
    const float* __restrict__ cand_val, const int* __restrict__ cand_idx,
    const float* __restrict__ pq_g, const float* __restrict__ value,
    float* __restrict__ out, int ncand) {

    __shared__ float cv[NB_SIM * KTOP];   // 20 KB
    __shared__ int   ci[NB_SIM * KTOP];   // 20 KB
    __shared__ float rv[256];
    __shared__ int   rp[256];
    __shared__ float topv[KTOP];
    __shared__ int   topi[KTOP];
    __shared__ float attn[KTOP];

    for (int i = threadIdx.x; i < ncand; i += 256) {
        cv[i] = cand_val[i];
        ci[i] = cand_idx[i];
    }
    __syncthreads();

    for (int sel = 0; sel < KTOP; ++sel) {
        float bv = -__builtin_inff(); int bp = 0;
        for (int i = threadIdx.x; i < ncand; i += 256) {
            const float v = cv[i];
            if (v > bv) { bv = v; bp = i; }
        }
        rv[threadIdx.x] = bv; rp[threadIdx.x] = bp;
        __syncthreads();
        for (int s = 128; s > 0; s >>= 1) {
            if (threadIdx.x < s) {
                if (rv[threadIdx.x + s] > rv[threadIdx.x]) {
                    rv[threadIdx.x] = rv[threadIdx.x + s];
                    rp[threadIdx.x] = rp[threadIdx.x + s];
                }
            }
            __syncthreads();
        }
        if (threadIdx.x == 0) {
            topv[sel] = rv[0];
            topi[sel] = ci[rp[0]];
            cv[rp[0]] = -__builtin_inff();
        }
        __syncthreads();
    }

    // softmax over the 10 selected scores (everything else is -inf in ref)
    if (threadIdx.x == 0) {
        const float m = topv[0];              // sorted descending -> max
        float e[KTOP]; float s = 0.f;
        #pragma unroll
        for (int i = 0; i < KTOP; ++i) { e[i] = expf(topv[i] - m); s += e[i]; }
        #pragma unroll
        for (int i = 0; i < KTOP; ++i) attn[i] = e[i] / s;
    }
    __syncthreads();

    // mastery[g] = sum_i attn[i] * value[idx_i][g];  logit = dot(pq, mastery)
    float partial = 0.f;
    if (threadIdx.x < DIM_G) {
        float mast = 0.f;
        #pragma unroll
        for (int i = 0; i < KTOP; ++i)
            mast = fmaf(attn[i], value[(size_t)topi[i] * DIM_G + threadIdx.x], mast);
        partial = mast * pq_g[threadIdx.x];
    }
    rv[threadIdx.x] = partial;
    __syncthreads();
    for (int s = 128; s > 0; s >>= 1) {
        if (threadIdx.x < s) rv[threadIdx.x] += rv[threadIdx.x + s];
        __syncthreads();
    }
    if (threadIdx.x == 0) out[0] = 1.f / (1.f + expf(-rv[0]));
}

// ---------------------------------------------------------------------------
extern "C" void kernel_launch(void* const* d_in, const int* in_sizes, int n_in,
                              void* d_out, int out_size, void* d_ws, size_t ws_size,
                              hipStream_t stream) {
    const float* qt  = (const float*)d_in[0];   // [1,512]
    const float* key = (const float*)d_in[1];   // [M,128]
    const float* val = (const float*)d_in[2];   // [M,128]
    const float* Wq  = (const float*)d_in[3];   // [128,512]
    const float* bq  = (const float*)d_in[4];   // [128]
    const int M = in_sizes[1] / DIM_G;

    float* wsf      = (float*)d_ws;
    float* pq       = wsf;                               // 128 floats
    float* cand_val = wsf + DIM_G;                       // NB_SIM*10 floats
    int*   cand_idx = (int*)(wsf + DIM_G + NB_SIM * KTOP);

    pq_kernel<<<1, 128, 0, stream>>>(qt, Wq, bq, pq);
    sim_topk_kernel<<<NB_SIM, 256, 0, stream>>>(key, pq, cand_val, cand_idx, M);
    final_kernel<<<1, 256, 0, stream>>>(cand_val, cand_idx, pq, val,
                                        (float*)d_out, NB_SIM * KTOP);
}